// Sopa_18897856102689
// MI455X (gfx1250) — compile-verified
//
#include <hip/hip_runtime.h>
#include <math.h>

// Problem constants (from reference setup_inputs)
#define L_DIM 512
#define B_DIM 64
#define D_DIM 256
#define N_PAT 256
#define P_LEN 6
#define E_DIM (N_PAT * 2 * P_LEN)   // 3072 transition columns
#define ZERO_MP (-100.0f)           // MaxPlus semiring zero

// GEMM block tiling: 8 waves = 2 (M) x 4 (N) wave-tiles of 16x16
#define BM 32
#define BN 64
#define KT 64
#define LDS_STRIDE 68               // 64 + 4 pad: rows 16B-aligned, conflict-free b64 reads
#define NPANELS (D_DIM / KT)        // 4

typedef float v2f __attribute__((ext_vector_type(2)));
typedef float v8f __attribute__((ext_vector_type(8)));

// Low 32 bits of a generic pointer to __shared__ = workgroup-relative LDS byte offset
__device__ __forceinline__ unsigned lds_off(const void* p) {
  return (unsigned)(unsigned long long)p;
}

// CDNA5 async copy global -> LDS (ASYNCcnt-tracked), GVS addressing:
//   LDS byte address from the vdst VGPR; global address = SGPR base + VGPR byte offset
__device__ __forceinline__ void async_b128(unsigned lds_byte, unsigned gl_byte,
                                           const float* base) {
  asm volatile("global_load_async_to_lds_b128 %0, %1, %2"
               :: "v"(lds_byte), "v"(gl_byte), "s"(base)
               : "memory");
}

// ---------------------------------------------------------------------------
// Kernel 1: tr[l,b,e] = x[l,b,:] . diags[e,:] + bias[e]  for l in [l0,l0+chunkL)
// GEMM M=(chunkL*B) x K=256 x N=3072 via V_WMMA_F32_16X16X4_F32,
// K-panels double-buffered in LDS via global_load_async_to_lds_b128.
//
// fp32 WMMA lane layout (CDNA5 ISA 7.12.2, 32-bit 16x4 A):
//   A: lane[0..15] -> M=lane, VGPR{0,1}=K{0,1}; lane[16..31] -> K{2,3}
//   B (4x16, mirrored): lane&15 -> N column, halves pick K{0,1}/{2,3}
//   C/D: VGPR r -> M = r + (lane<16 ? 0 : 8), N = lane&15
// ---------------------------------------------------------------------------
__global__ __launch_bounds__(256) void sopa_gemm_kernel(
    const float* __restrict__ x, const float* __restrict__ diags,
    const float* __restrict__ bias, float* __restrict__ tr,
    int l0, int chunkL)
{
  __shared__ float xs[2][BM][LDS_STRIDE];   // X panel:      32 rows x 64 k
  __shared__ float dsb[2][BN][LDS_STRIDE];  // diags panel:  64 cols x 64 k

  const int tid  = threadIdx.x;
  const int lane = tid & 31;
  const int wave = tid >> 5;

  const int nBlkN = E_DIM / BN;             // 48
  const int mb = blockIdx.x / nBlkN;
  const int nb = blockIdx.x - mb * nBlkN;
  const int m0 = mb * BM;                   // chunk-relative output row base
  const int n0 = nb * BN;                   // output column base

  const int r15  = lane & 15;
  const int hi   = lane >> 4;
  const int koff = hi << 1;                 // 0 or 2

  const int mloc = (wave >> 2) << 4;        // 0 / 16
  const int nloc = (wave & 3) << 4;         // 0 / 16 / 32 / 48

  // Stage one K-panel (6 x b128 async ops per thread: 2 for X, 4 for diags)
  auto issue_panel = [&](int buf, int k0) {
#pragma unroll
    for (int j = 0; j < 2; ++j) {           // X: 32 rows x 16 chunks = 512
      int c = tid + 256 * j;
      int row = c >> 4, q = c & 15;
      unsigned l = lds_off(&xs[buf][row][q * 4]);
      unsigned g = (unsigned)(((l0 * B_DIM + m0 + row) * D_DIM + k0 + q * 4) * 4);
      async_b128(l, g, x);
    }
#pragma unroll
    for (int j = 0; j < 4; ++j) {           // diags: 64 rows x 16 chunks = 1024
      int c = tid + 256 * j;
      int row = c >> 4, q = c & 15;
      unsigned l = lds_off(&dsb[buf][row][q * 4]);
      unsigned g = (unsigned)(((n0 + row) * D_DIM + k0 + q * 4) * 4);
      async_b128(l, g, diags);
    }
  };

  v8f acc = {};
  issue_panel(0, 0);

  for (int p = 0; p < NPANELS; ++p) {
    if (p + 1 < NPANELS) {
      issue_panel((p + 1) & 1, (p + 1) * KT);
      asm volatile("s_wait_asynccnt 0x6" ::: "memory");  // panel p's 6 done
    } else {
      asm volatile("s_wait_asynccnt 0x0" ::: "memory");  // drain
    }
    __syncthreads();                        // all waves' LDS writes visible

    const float* xp = &xs[p & 1][mloc + r15][koff];
    const float* dp = &dsb[p & 1][nloc + r15][koff];
#pragma unroll
    for (int kk = 0; kk < KT; kk += 4) {
      v2f a = *(const v2f*)(xp + kk);       // ds_load_b64, conflict-free
      v2f b = *(const v2f*)(dp + kk);
      acc = __builtin_amdgcn_wmma_f32_16x16x4_f32(
          /*neg_a=*/false, a, /*neg_b=*/false, b,
          /*c_mod=*/(short)0, acc, /*reuse_a=*/false, /*reuse_b=*/false);
    }
    __syncthreads();                        // panel consumed before overwrite
  }

  const int col = n0 + nloc + r15;
  const float bv = bias[col];
  const int mbase = m0 + mloc + (hi << 3);  // +8 for lanes 16..31
#pragma unroll
  for (int r = 0; r < 8; ++r) {
    tr[(size_t)(mbase + r) * E_DIM + col] = acc[r] + bv;
  }
}

// ---------------------------------------------------------------------------
// Kernel 2: sequential max-plus scan for l in [l0, l0+chunkL).
// One thread per (b, n) pattern; h[P] in registers; tr read as 3x float4.
// tr layout per (l,b): e = n*12 + sigma*6 + p  (sigma 0 = self-loop, 1 = main)
// ---------------------------------------------------------------------------
__global__ __launch_bounds__(256) void sopa_scan_kernel(
    const float* __restrict__ tr, const float* __restrict__ eps,
    const int* __restrict__ input_len, float* __restrict__ out,
    float* __restrict__ hstate, float* __restrict__ sstate,
    int l0, int chunkL)
{
  const int t = blockIdx.x * 256 + threadIdx.x;   // [0, B*N)
  const int b = t >> 8;
  const int n = t & (N_PAT - 1);

  float h[P_LEN];
  float s;
  if (l0 == 0) {
    h[0] = 0.0f;                                  // semiring.one at state 0
#pragma unroll
    for (int p = 1; p < P_LEN; ++p) h[p] = ZERO_MP;
    s = ZERO_MP;
  } else {
#pragma unroll
    for (int p = 0; p < P_LEN; ++p) h[p] = hstate[(size_t)t * P_LEN + p];
    s = sstate[t];
  }

  float e[P_LEN - 1];
#pragma unroll
  for (int q = 0; q < P_LEN - 1; ++q) e[q] = eps[n * (P_LEN - 1) + q];

  const int es = 2 + (n >> 6);                    // end state: plen-1, plen=3..6
  const int len = input_len[b];

  const char* base = (const char*)(tr + (size_t)b * E_DIM + (size_t)n * 12);
  const size_t lstride = (size_t)B_DIM * E_DIM * sizeof(float);

  for (int lr = 0; lr < chunkL; ++lr) {
    const int l = l0 + lr;
    const float4* p4 = (const float4*)(base + (size_t)lr * lstride);
    float4 v0 = p4[0];
    float4 v1 = p4[1];
    float4 v2 = p4[2];
    const float tr0[P_LEN] = {v0.x, v0.y, v0.z, v0.w, v1.x, v1.y};  // self-loop
    const float tr1[P_LEN] = {v1.z, v1.w, v2.x, v2.y, v2.z, v2.w};  // main path

    float ae[P_LEN];
    ae[0] = fmaxf(h[0], ZERO_MP);
#pragma unroll
    for (int p = 1; p < P_LEN; ++p) ae[p] = fmaxf(h[p], h[p - 1] + e[p - 1]);

    h[0] = fmaxf(0.0f, ae[0] + tr0[0]);           // one_pad vs self-loop
#pragma unroll
    for (int p = 1; p < P_LEN; ++p)
      h[p] = fmaxf(ae[p - 1] + tr1[p - 1], ae[p] + tr0[p]);

    const float ev = (es == 2) ? h[2] : (es == 3) ? h[3] : (es == 4) ? h[4] : h[5];
    if (len >= l) s = fmaxf(s, ev);
    out[(size_t)l * (B_DIM * N_PAT) + t] = tanhf(s);
  }

#pragma unroll
  for (int p = 0; p < P_LEN; ++p) hstate[(size_t)t * P_LEN + p] = h[p];
  sstate[t] = s;
}

// ---------------------------------------------------------------------------
// Launch: chunk L so the tr intermediate stays L2-resident (<=50MB @ chunk 64).
// Chunk size chosen deterministically from ws_size.
// ---------------------------------------------------------------------------
extern "C" void kernel_launch(void* const* d_in, const int* in_sizes, int n_in,
                              void* d_out, int out_size, void* d_ws, size_t ws_size,
                              hipStream_t stream)
{
  const float* x         = (const float*)d_in[0];   // (L,B,D)
  const int*   input_len = (const int*)d_in[1];     // (B,)
  const float* diags     = (const float*)d_in[2];   // (2NP, D)
  const float* bias      = (const float*)d_in[3];   // (2NP, 1)
  const float* eps       = (const float*)d_in[4];   // (N, P-1)
  float* out = (float*)d_out;                       // (L,B,N)

  float* hstate = (float*)d_ws;                                   // B*N*P
  float* sstate = hstate + (size_t)B_DIM * N_PAT * P_LEN;         // B*N
  float* trws   = sstate + (size_t)B_DIM * N_PAT;                 // chunk*B*E
  const size_t stateBytes = (size_t)B_DIM * N_PAT * (P_LEN + 1) * sizeof(float);

  int chunk = 1;
  for (int c = 64; c >= 1; c >>= 1) {
    if (stateBytes + (size_t)c * B_DIM * E_DIM * sizeof(float) <= ws_size) {
      chunk = c;
      break;
    }
  }

  for (int l0 = 0; l0 < L_DIM; l0 += chunk) {
    const int cl = (L_DIM - l0 < chunk) ? (L_DIM - l0) : chunk;
    const int blocks = ((cl * B_DIM) / BM) * (E_DIM / BN);
    sopa_gemm_kernel<<<blocks, 256, 0, stream>>>(
        x, diags, bias, trws, l0, cl);
    sopa_scan_kernel<<<(B_DIM * N_PAT) / 256, 256, 0, stream>>>(
        trws, eps, input_len, out, hstate, sstate, l0, cl);
  }
}